// HyperConv_78572131713073
// MI455X (gfx1250) — compile-verified
//
#include <hip/hip_runtime.h>

#define EMB 100
#define TILE_EDGES 16

typedef __attribute__((ext_vector_type(2))) float v2f;
typedef __attribute__((ext_vector_type(8))) float v8f;

// ---------------------------------------------------------------------------
// Elementwise helpers
// ---------------------------------------------------------------------------
__global__ __launch_bounds__(256) void hc_zero(float* __restrict__ p, int n) {
  int i = blockIdx.x * blockDim.x + threadIdx.x;
  if (i < n) p[i] = 0.0f;
}

__global__ __launch_bounds__(256) void hc_copy(float* __restrict__ o,
                                               const float* __restrict__ a, int n) {
  int i = blockIdx.x * blockDim.x + threadIdx.x;
  if (i < n) o[i] = a[i];
}

__global__ __launch_bounds__(256) void hc_add(float* __restrict__ o,
                                              const float* __restrict__ a, int n) {
  int i = blockIdx.x * blockDim.x + threadIdx.x;
  if (i < n) o[i] += a[i];
}

__global__ __launch_bounds__(256) void hc_add_scale(float* __restrict__ o,
                                                    const float* __restrict__ a,
                                                    float s, int n) {
  int i = blockIdx.x * blockDim.x + threadIdx.x;
  if (i < n) o[i] = (o[i] + a[i]) * s;
}

// ---------------------------------------------------------------------------
// WMMA-tiled atomic-scatter SpMM: one wave32 per 16-edge tile.
//   D(16x16) = diag(vals[tile]) * X_gathered(16 x 16-col-chunk), via 4x
//   V_WMMA_F32_16X16X4_F32 accumulations (full fp32, matches reference).
//   Each lane then scatters its 8 D elements (8 rows, 1 column) with
//   native fp32 global atomics (no-return -> STOREcnt path, L2 RMW).
// ---------------------------------------------------------------------------
__global__ __launch_bounds__(256) void hc_spmm_wmma(const int*   __restrict__ rows,
                                                    const int*   __restrict__ cols,
                                                    const float* __restrict__ vals,
                                                    const float* __restrict__ x,
                                                    float*       __restrict__ y,
                                                    int n_tiles) {
  const int lane = threadIdx.x & 31;
  const int tile = blockIdx.x * (blockDim.x >> 5) + (threadIdx.x >> 5);
  if (tile >= n_tiles) return;              // wave-uniform: EXEC stays all-1s
  const int ebase = tile * TILE_EDGES;
  const int m     = lane & 15;              // matrix row (A) / column (B,D)
  const int half  = lane >> 4;              // lane-half selects K / M ranges

  // Diagonal A value for row m (broadcast-friendly 64B load).
  const float aval = vals[ebase + m];

  // Prefetch edge data a few tiles ahead (global_prefetch_b8, L2 resident).
  __builtin_prefetch(&rows[ebase + 8 * TILE_EDGES], 0, 0);
  __builtin_prefetch(&cols[ebase + 8 * TILE_EDGES], 0, 0);
  __builtin_prefetch(&vals[ebase + 8 * TILE_EDGES], 0, 0);

  // Destination rows for the scatter phase: lane needs rows M = r + 8*half.
  int drow[8];
#pragma unroll
  for (int r = 0; r < 8; ++r)
    drow[r] = rows[ebase + r + 8 * half] * EMB;

  // Source rows for the gather phase. For chunk k, B-VGPR r holds
  // K = 2*half + r, i.e. edge index 4k + 2*half + r (uniform per lane-half).
  int scol[4][2];
#pragma unroll
  for (int k = 0; k < 4; ++k)
#pragma unroll
    for (int r = 0; r < 2; ++r)
      scol[k][r] = cols[ebase + 4 * k + 2 * half + r] * EMB;

  for (int cb = 0; cb < EMB; cb += 16) {
    const int  col   = cb + m;
    const bool valid = (col < EMB);         // last chunk covers cols 96..99

    v8f c = {0.f, 0.f, 0.f, 0.f, 0.f, 0.f, 0.f, 0.f};
#pragma unroll
    for (int k = 0; k < 4; ++k) {
      v2f a, b;
#pragma unroll
      for (int r = 0; r < 2; ++r) {
        const int kk = 2 * half + r;        // K index held in this VGPR/lane
        // A 16x4 chunk k of diag(vals): nonzero only where m == 4k + kk
        a[r] = (m == 4 * k + kk) ? aval : 0.0f;
        // B 4x16 chunk k: row K = x[cols[4k+kk]], columns cb..cb+15
        b[r] = valid ? x[scol[k][r] + col] : 0.0f;
      }
      c = __builtin_amdgcn_wmma_f32_16x16x4_f32(
              /*neg_a=*/false, a, /*neg_b=*/false, b,
              /*c_mod=*/(short)0, c, /*reuse_a=*/false, /*reuse_b=*/false);
    }

    if (valid) {
#pragma unroll
      for (int r = 0; r < 8; ++r)
        unsafeAtomicAdd(&y[drow[r] + col], c[r]);   // global_atomic_add_f32
    }
  }
}

// Scalar tail for edge counts not divisible by 16 (none for E=800000).
__global__ __launch_bounds__(256) void hc_spmm_tail(const int*   __restrict__ rows,
                                                    const int*   __restrict__ cols,
                                                    const float* __restrict__ vals,
                                                    const float* __restrict__ x,
                                                    float*       __restrict__ y,
                                                    int estart, int E) {
  int t = blockIdx.x * blockDim.x + threadIdx.x;
  int e = estart + t / EMB;
  int d = t % EMB;
  if (e < E)
    unsafeAtomicAdd(&y[rows[e] * EMB + d], vals[e] * x[cols[e] * EMB + d]);
}

// ---------------------------------------------------------------------------
// Host-side orchestration (graph-capture safe: launches only)
// ---------------------------------------------------------------------------
extern "C" void kernel_launch(void* const* d_in, const int* in_sizes, int n_in,
                              void* d_out, int out_size, void* d_ws, size_t ws_size,
                              hipStream_t stream) {
  const int*   rows = (const int*)d_in[0];
  const int*   cols = (const int*)d_in[1];
  const float* vals = (const float*)d_in[2];
  const float* emb  = (const float*)d_in[3];

  const int E = in_sizes[0];
  const int n = in_sizes[3];                 // N_ITEMS * EMB flat elements
  float* out = (float*)d_out;
  float* A   = (float*)d_ws;                 // 2 x (n floats) ping-pong scratch
  float* B   = A + n;

  const int tpb = 256;
  const int eg  = (n + tpb - 1) / tpb;

  const int n_tiles   = E / TILE_EDGES;
  const int rem_start = n_tiles * TILE_EDGES;
  const int waves_pb  = tpb / 32;
  const int sg        = (n_tiles + waves_pb - 1) / waves_pb;
  const int tail_thr  = (E - rem_start) * EMB;
  const int tail_g    = (tail_thr + tpb - 1) / tpb;

  // hop-0 term
  hc_copy<<<eg, tpb, 0, stream>>>(out, emb, n);

  const float* xin = emb;
  float* bufs[3] = {A, B, A};                // A already consumed before reuse
  for (int layer = 0; layer < 3; ++layer) {
    float* y = bufs[layer];
    hc_zero<<<eg, tpb, 0, stream>>>(y, n);
    hc_spmm_wmma<<<sg, tpb, 0, stream>>>(rows, cols, vals, xin, y, n_tiles);
    if (tail_thr > 0)
      hc_spmm_tail<<<tail_g, tpb, 0, stream>>>(rows, cols, vals, xin, y, rem_start, E);
    if (layer < 2)
      hc_add<<<eg, tpb, 0, stream>>>(out, y, n);
    else
      hc_add_scale<<<eg, tpb, 0, stream>>>(out, y, 0.25f, n);
    xin = y;
  }
}